// CVXPolicy_DoubleIntegrator_14216341749948
// MI455X (gfx1250) — compile-verified
//
#include <hip/hip_runtime.h>
#include <hip/hip_bf16.h>
#include <math.h>

// -------- CDNA5 (gfx1250) wave32 WMMA fp32 pipeline, persistent WGs --------
typedef float v2f __attribute__((ext_vector_type(2)));
typedef float v8f __attribute__((ext_vector_type(8)));

#define LDS_FENCE()  asm volatile("s_wait_dscnt 0x0" ::: "memory")
#define ASYNC_WAIT() asm volatile("s_wait_asynccnt 0x0" ::: "memory")

// Async DMA: 16 bytes global -> LDS per active lane (ASYNCcnt-tracked)
__device__ __forceinline__ void async_ldg_b128(unsigned lds_byte_off, const float* g) {
    asm volatile("global_load_async_to_lds_b128 %0, %1, off"
                 :: "v"(lds_byte_off), "v"((unsigned long long)(uintptr_t)g)
                 : "memory");
}

// exp-based tanh using HW v_exp_f32; saturates for large |x|
__device__ __forceinline__ float fast_tanh(float x) {
    float e = __expf(-2.0f * fabsf(x));
    float r = __fdividef(1.0f - e, 1.0f + e);
    return copysignf(r, x);
}

// Problem constants
#define IN_DIM   193   // 1 + 192 (X re-ordered as [z(0..191) | t(192)], W1 rows permuted to match)
#define IN_PAD   196   // 49 K-steps of 4
#define HID      100
#define HID_PAD  112   // 7 N-tiles of 16
#define OUTD     192   // 12 N-tiles of 16
#define W3_PAD   208   // bank-padded N stride for W3 (2*208 % 64 == 32)
#define HS_STR   116   // bank-friendly h-tile row stride
#define CTRL     96
#define GRIDMAX  512   // persistent workgroups

__global__ __launch_bounds__(128)
void cvx_policy_kernel(const float* __restrict__ z,  const float* __restrict__ t,
                       const float* __restrict__ W1, const float* __restrict__ b1,
                       const float* __restrict__ W2, const float* __restrict__ b2,
                       const float* __restrict__ W3, const float* __restrict__ b3,
                       float* __restrict__ out, int nblk)
{
    // Weights staged once per WG, K-pair interleaved:
    //   Ws[(pair*STRIDE + col)*2 + (K&1)], pair = K>>1
    // => a WMMA B-fragment (two adjacent K rows, one col) is one aligned b64.
    __shared__ float W1s[(IN_PAD/2) * HID_PAD * 2];   // 98 pairs x 112
    __shared__ float W2s[(HID/2)   * HID_PAD * 2];    // 50 pairs x 112
    __shared__ float W3s[(HID/2)   * W3_PAD * 2];     // 50 pairs x 208
    __shared__ float b1s[HID_PAD], b2s[HID_PAD], b3s[OUTD];
    __shared__ float xbuf[4][16 * IN_PAD];            // X tile, reused as p tile (16x192)
    __shared__ float hbuf[4][16 * HS_STR];            // h1/h2 tile, reused as q tile (16x96)
    __shared__ float rbuf[4][16];
    __shared__ float sbuf[4][16];

    const int tid  = threadIdx.x;
    const int wave = tid >> 5;
    const int lane = tid & 31;
    const int half = lane >> 4;   // 0: K rows 4k,4k+1 ; 1: K rows 4k+2,4k+3
    const int mrow = lane & 15;

    float* xs = xbuf[wave];
    float* hs = hbuf[wave];
    const unsigned xs_base = (unsigned)(uintptr_t)xs;  // flat LDS addr low bits = LDS offset

    // Stage one X = [z | t] tile (16 x 196) for row block `blk` via async DMA
    auto issue_x = [&](int blk) {
        const int row0 = (blk * 4 + wave) * 16;
        const float* zb = z + (size_t)row0 * 192;
        #pragma unroll 4
        for (int i = lane; i < 16 * 48; i += 32) {            // 48 x b128 chunks per row
            int r = i / 48, c = i % 48;
            async_ldg_b128(xs_base + (unsigned)(r * IN_PAD + c * 4) * 4u, zb + r * 192 + c * 4);
        }
        if (lane < 16) xs[lane * IN_PAD + 192] = t[row0 + lane];
        for (int i = lane; i < 16 * 3; i += 32)               // zero cols 193..195
            xs[(i / 3) * IN_PAD + 193 + (i % 3)] = 0.0f;
    };

    int blk = blockIdx.x;
    if (blk < nblk) issue_x(blk);       // overlap first X load with weight staging

    // ---------------- cooperative weight staging (swizzled, once per WG) ----------------
    for (int i = tid; i < (IN_PAD/2) * HID_PAD * 2; i += 128) W1s[i] = 0.0f;
    for (int i = tid; i < (HID/2)   * HID_PAD * 2; i += 128) W2s[i] = 0.0f;
    __syncthreads();
    for (int i = tid; i < IN_DIM * HID; i += 128) {
        int r = i / HID, c = i % HID;
        int nr = (r == 0) ? 192 : (r - 1);                    // [z | t] row permutation
        W1s[((nr >> 1) * HID_PAD + c) * 2 + (nr & 1)] = W1[i];
    }
    for (int i = tid; i < HID * HID; i += 128) {
        int r = i / HID, c = i % HID;
        W2s[((r >> 1) * HID_PAD + c) * 2 + (r & 1)] = W2[i];
    }
    for (int i = tid; i < HID * OUTD; i += 128) {
        int r = i / OUTD, c = i % OUTD;
        W3s[((r >> 1) * W3_PAD + c) * 2 + (r & 1)] = W3[i];
    }
    if (tid < HID_PAD) {
        b1s[tid] = (tid < HID) ? b1[tid] : 0.0f;
        b2s[tid] = (tid < HID) ? b2[tid] : 0.0f;
    }
    if (tid < OUTD) b3s[tid] = b3[tid];
    __syncthreads();

    const v8f vzero = {0.f,0.f,0.f,0.f,0.f,0.f,0.f,0.f};
    const int j0 = lane, j1 = lane + 32, j2 = lane + 64;

    for (; blk < nblk; blk += gridDim.x) {
        const int row0 = (blk * 4 + wave) * 16;
        ASYNC_WAIT();                   // X tile resident
        LDS_FENCE();

        // ---------------- GEMM1: h1[16x100] = X[16x193] * W1 ----------------
        v8f acc1[7];
        #pragma unroll
        for (int n = 0; n < 7; ++n) acc1[n] = vzero;
        {
            const float* xrow = xs + mrow * IN_PAD + 2 * half;
            for (int k = 0; k < 49; ++k) {
                v2f a = *(const v2f*)(xrow + 4 * k);
                const float* w = W1s + ((2 * k + half) * HID_PAD + mrow) * 2;
                #pragma unroll
                for (int n = 0; n < 7; ++n) {
                    v2f b = *(const v2f*)(w + 32 * n);
                    acc1[n] = __builtin_amdgcn_wmma_f32_16x16x4_f32(
                        false, a, false, b, (short)0, acc1[n], false, false);
                }
            }
        }
        #pragma unroll
        for (int n = 0; n < 7; ++n) {
            float bb = b1s[16 * n + mrow];
            #pragma unroll
            for (int j = 0; j < 8; ++j)
                hs[(j + 8 * half) * HS_STR + 16 * n + mrow] = fast_tanh(acc1[n][j] + bb);
        }
        LDS_FENCE();

        // ---------------- GEMM2: h2[16x100] = h1 * W2 ----------------
        v8f acc2[7];
        #pragma unroll
        for (int n = 0; n < 7; ++n) acc2[n] = vzero;
        {
            const float* hrow = hs + mrow * HS_STR + 2 * half;
            for (int k = 0; k < 25; ++k) {
                v2f a = *(const v2f*)(hrow + 4 * k);
                const float* w = W2s + ((2 * k + half) * HID_PAD + mrow) * 2;
                #pragma unroll
                for (int n = 0; n < 7; ++n) {
                    v2f b = *(const v2f*)(w + 32 * n);
                    acc2[n] = __builtin_amdgcn_wmma_f32_16x16x4_f32(
                        false, a, false, b, (short)0, acc2[n], false, false);
                }
            }
        }
        LDS_FENCE();
        #pragma unroll
        for (int n = 0; n < 7; ++n) {
            float bb = b2s[16 * n + mrow];
            #pragma unroll
            for (int j = 0; j < 8; ++j)
                hs[(j + 8 * half) * HS_STR + 16 * n + mrow] = fast_tanh(acc2[n][j] + bb);
        }
        LDS_FENCE();

        // ---------------- GEMM3: p[16x192] = h2 * W3 ----------------
        v8f acc3[12];
        #pragma unroll
        for (int n = 0; n < 12; ++n) acc3[n] = vzero;
        {
            const float* hrow = hs + mrow * HS_STR + 2 * half;
            for (int k = 0; k < 25; ++k) {
                v2f a = *(const v2f*)(hrow + 4 * k);
                const float* w = W3s + ((2 * k + half) * W3_PAD + mrow) * 2;
                #pragma unroll
                for (int n = 0; n < 12; ++n) {
                    v2f b = *(const v2f*)(w + 32 * n);
                    acc3[n] = __builtin_amdgcn_wmma_f32_16x16x4_f32(
                        false, a, false, b, (short)0, acc3[n], false, false);
                }
            }
        }
        float* ps = xs;                 // reuse X-tile LDS for p (16 x 192)
        #pragma unroll
        for (int n = 0; n < 12; ++n) {
            float bb = b3s[16 * n + mrow];
            #pragma unroll
            for (int j = 0; j < 8; ++j)
                ps[(j + 8 * half) * OUTD + 16 * n + mrow] = acc3[n][j] + bb;
        }
        LDS_FENCE();

        // ---------------- scatter p -> q, r = ||q||^2 per row ----------------
        // q[j] = (j<32 ? p[3j+3] : 0) + (j<64 && j even ? p[2j+4] : 0) + (j%3==0 ? p[5*(j/3)+5] : 0)
        float* qs = hs;                 // reuse h-tile LDS for q (16 x 96)
        for (int row = 0; row < 16; ++row) {
            const float* pr = ps + row * OUTD;
            float q0 = pr[3 * j0 + 3];
            q0 += ((j0 & 1) == 0) ? pr[2 * j0 + 4] : 0.0f;
            q0 += ((j0 % 3) == 0) ? pr[5 * (j0 / 3) + 5] : 0.0f;
            float q1 = ((j1 & 1) == 0) ? pr[2 * j1 + 4] : 0.0f;
            q1 += ((j1 % 3) == 0) ? pr[5 * (j1 / 3) + 5] : 0.0f;
            float q2 = ((j2 % 3) == 0) ? pr[5 * (j2 / 3) + 5] : 0.0f;
            qs[row * CTRL + j0] = q0;
            qs[row * CTRL + j1] = q1;
            qs[row * CTRL + j2] = q2;
            float rr = q0 * q0 + q1 * q1 + q2 * q2;
            #pragma unroll
            for (int off = 16; off > 0; off >>= 1)
                rr += __shfl_xor(rr, off, 32);
            if (lane == 0) rbuf[wave][row] = rr;
        }
        LDS_FENCE();                    // all ps reads retired before async overwrite of xs

        // Prefetch next block's X tile; latency hides under Newton + stores
        if (blk + (int)gridDim.x < nblk) issue_x(blk + gridDim.x);

        // ---------------- Newton: 16 rows in parallel across lanes 0..15 ----------------
        if (lane < 16) {
            float rr = rbuf[wave][lane];
            float s = cbrtf(rr);
            #pragma unroll 1
            for (int it = 0; it < 40; ++it) {
                float one = 1.0f + s;
                float f  = s * one * one - rr;
                float fp = one * one + 2.0f * s * one;
                s = fmaxf(s - f / fp, 0.0f);
            }
            sbuf[wave][lane] = 1.0f / (1.0f + s);
        }
        LDS_FENCE();

        // ---------------- u = -q / (1+s), streaming (non-temporal) store ----------------
        for (int row = 0; row < 16; ++row) {
            float inv = sbuf[wave][row];
            float* orow = out + (size_t)(row0 + row) * CTRL;
            __builtin_nontemporal_store(-qs[row * CTRL + j0] * inv, orow + j0);
            __builtin_nontemporal_store(-qs[row * CTRL + j1] * inv, orow + j1);
            __builtin_nontemporal_store(-qs[row * CTRL + j2] * inv, orow + j2);
        }
        LDS_FENCE();                    // qs reads retired before next iteration reuses hs
    }
}

extern "C" void kernel_launch(void* const* d_in, const int* in_sizes, int n_in,
                              void* d_out, int out_size, void* d_ws, size_t ws_size,
                              hipStream_t stream) {
    const float* z  = (const float*)d_in[0];
    const float* t  = (const float*)d_in[1];
    const float* W1 = (const float*)d_in[2];
    const float* b1 = (const float*)d_in[3];
    const float* W2 = (const float*)d_in[4];
    const float* b2 = (const float*)d_in[5];
    const float* W3 = (const float*)d_in[6];
    const float* b3 = (const float*)d_in[7];
    float* out = (float*)d_out;

    const int B    = in_sizes[1];       // t is [B,1]
    const int nblk = B / 64;            // 4 waves/WG x 16 rows/wave
    const int grid = (nblk < GRIDMAX) ? nblk : GRIDMAX;
    cvx_policy_kernel<<<grid, 128, 0, stream>>>(z, t, W1, b1, W2, b2, W3, b3, out, nblk);
}